// GAT0_23931557773765
// MI455X (gfx1250) — compile-verified
//
#include <hip/hip_runtime.h>
#include <math.h>

#define N_NODES 50000
#define E_EDGES 800000
#define E_TOT   850000          // edges + self loops
#define HEADS   3
#define CH      32              // per-head channels (HID == OUT == 32)
#define FDIM    96              // HEADS*CH
#define NEG_SLOPE 0.2f

typedef __attribute__((ext_vector_type(2))) float v2f;
typedef __attribute__((ext_vector_type(8))) float v8f;

// ---------------------------------------------------------------------------
// C[M,Nc] = A[M,K] @ B[K,Nc], fp32, one wave per 16x16 tile via
// V_WMMA_F32_16X16X4_F32.  M%16==0, Nc%16==0, K%4==0 guaranteed by caller.
// A 16x4 layout: lanes 0-15 = rows, VGPR0/1 = K {0,1} (half 0) / {2,3} (half 1)
// B 4x16 layout: lanes = cols, VGPR0/1 = K rows {0,1}/{2,3} mirrored
// C/D: lane l<16 -> N=l, M=v (+8 for upper half-lanes)
// ---------------------------------------------------------------------------
__global__ void gemm_wmma(const float* __restrict__ A, const float* __restrict__ B,
                          float* __restrict__ C, int M, int K, int Nc) {
  const int wavesPerBlock = blockDim.x >> 5;
  const int wave = blockIdx.x * wavesPerBlock + (threadIdx.x >> 5);
  const int colTiles = Nc >> 4;
  const int rowTile = wave / colTiles;
  const int colTile = wave - rowTile * colTiles;
  if (rowTile * 16 >= M) return;                 // wave-uniform: EXEC stays all-1s
  const int lane = threadIdx.x & 31;
  const int half = lane >> 4;
  const int l    = lane & 15;
  const int row0 = rowTile << 4, col0 = colTile << 4;
  const float* arow = A + (size_t)(row0 + l) * K;
  v8f acc = {};
  for (int k = 0; k < K; k += 4) {
    v2f a, b;
    a.x = arow[k + 2 * half];
    a.y = arow[k + 2 * half + 1];
    b.x = B[(size_t)(k + 2 * half)     * Nc + col0 + l];
    b.y = B[(size_t)(k + 2 * half + 1) * Nc + col0 + l];
    acc = __builtin_amdgcn_wmma_f32_16x16x4_f32(false, a, false, b,
                                                (short)0, acc, false, false);
  }
  float* crow = C + (size_t)(row0 + 8 * half) * Nc + col0 + l;
#pragma unroll
  for (int v = 0; v < 8; ++v) crow[(size_t)v * Nc] = acc[v];
}

// es[n,h] = <h[n,h,:], a_src[h,:]>, ed likewise. One thread per (node, head).
__global__ void attn_scores(const float* __restrict__ hf, const float* __restrict__ a_src,
                            const float* __restrict__ a_dst, float* __restrict__ es,
                            float* __restrict__ ed) {
  int idx = blockIdx.x * blockDim.x + threadIdx.x;
  if (idx >= N_NODES * HEADS) return;
  int n = idx / HEADS, hd = idx - n * HEADS;
  const float* hp = hf + (size_t)n * FDIM + hd * CH;
  const float* sp = a_src + hd * CH;
  const float* dp = a_dst + hd * CH;
  float s = 0.f, d = 0.f;
#pragma unroll
  for (int c = 0; c < CH; ++c) { float v = hp[c]; s += v * sp[c]; d += v * dp[c]; }
  es[idx] = s; ed[idx] = d;
}

__global__ void seg_init(float* __restrict__ acc, float* __restrict__ amax,
                         float* __restrict__ den) {
  int i = blockIdx.x * blockDim.x + threadIdx.x;
  if (i < N_NODES * FDIM)  acc[i] = 0.f;
  if (i < N_NODES * HEADS) { amax[i] = -INFINITY; den[i] = 0.f; }
}

__device__ __forceinline__ void edge_ep(const long long* __restrict__ ei, int e,
                                        int& s, int& d) {
  if (e < E_EDGES) { s = (int)ei[e]; d = (int)ei[E_EDGES + e]; }
  else             { s = d = e - E_EDGES; }               // self loop
}

__device__ __forceinline__ void atomicMaxF(float* addr, float val) {
  if (val >= 0.f) atomicMax((int*)addr, __float_as_int(val));
  else            atomicMin((unsigned int*)addr, __float_as_uint(val));
}

// pass 1: alpha = leaky_relu(es[src]+ed[dst]); segment max over dst
__global__ void edge_alpha_max(const long long* __restrict__ ei,
                               const float* __restrict__ es, const float* __restrict__ ed,
                               float* __restrict__ alpha, float* __restrict__ amax) {
  int e = blockIdx.x * blockDim.x + threadIdx.x;
  if (e >= E_TOT) return;
  int s, d; edge_ep(ei, e, s, d);
#pragma unroll
  for (int h = 0; h < HEADS; ++h) {
    float a = es[s * HEADS + h] + ed[d * HEADS + h];
    a = (a > 0.f) ? a : NEG_SLOPE * a;
    alpha[(size_t)e * HEADS + h] = a;
    atomicMaxF(&amax[d * HEADS + h], a);
  }
}

// pass 2: ex = exp(alpha - amax[dst]); den[dst] += ex  (ex overwrites alpha)
__global__ void edge_exp_sum(const long long* __restrict__ ei,
                             const float* __restrict__ amax, float* __restrict__ alpha,
                             float* __restrict__ den) {
  int e = blockIdx.x * blockDim.x + threadIdx.x;
  if (e >= E_TOT) return;
  int s, d; edge_ep(ei, e, s, d);
#pragma unroll
  for (int h = 0; h < HEADS; ++h) {
    float ex = expf(alpha[(size_t)e * HEADS + h] - amax[d * HEADS + h]);
    alpha[(size_t)e * HEADS + h] = ex;
    atomicAdd(&den[d * HEADS + h], ex);
  }
}

// pass 3: acc[dst] += h[src] * att — one wave per edge, lane = channel,
// coalesced 128B gather + 128B atomic-add segments per head.
__global__ void edge_aggregate(const long long* __restrict__ ei,
                               const float* __restrict__ hf, const float* __restrict__ ex,
                               const float* __restrict__ den, float* __restrict__ acc) {
  int wave = blockIdx.x * (blockDim.x >> 5) + (threadIdx.x >> 5);
  if (wave >= E_TOT) return;
  int lane = threadIdx.x & 31;
  int s, d; edge_ep(ei, wave, s, d);
  const float* hsrc = hf + (size_t)s * FDIM;
  float* adst = acc + (size_t)d * FDIM;
#pragma unroll
  for (int i = 0; i < HEADS; ++i) {
    float att = ex[(size_t)wave * HEADS + i] / (den[d * HEADS + i] + 1e-16f);
    atomicAdd(&adst[i * CH + lane], hsrc[i * CH + lane] * att);
  }
}

__global__ void bias_relu(float* __restrict__ acc, const float* __restrict__ b) {
  int i = blockIdx.x * blockDim.x + threadIdx.x;
  if (i >= N_NODES * FDIM) return;
  float v = acc[i] + b[i % FDIM];
  acc[i] = v > 0.f ? v : 0.f;
}

__global__ void mean_bias_sigmoid(const float* __restrict__ acc,
                                  const float* __restrict__ b2, float* __restrict__ out) {
  int i = blockIdx.x * blockDim.x + threadIdx.x;
  if (i >= N_NODES * CH) return;
  int n = i / CH, c = i - n * CH;
  const float* a = acc + (size_t)n * FDIM;
  float v = (a[c] + a[CH + c] + a[2 * CH + c]) * (1.f / 3.f) + b2[c];
  out[i] = 1.f / (1.f + expf(-v));
}

extern "C" void kernel_launch(void* const* d_in, const int* in_sizes, int n_in,
                              void* d_out, int out_size, void* d_ws, size_t ws_size,
                              hipStream_t stream) {
  (void)in_sizes; (void)n_in; (void)out_size; (void)ws_size;
  const float*     x  = (const float*)d_in[0];
  const long long* ei = (const long long*)d_in[1];        // int64 edge_index [2,E]
  const float* W [3] = {(const float*)d_in[2], (const float*)d_in[6],  (const float*)d_in[10]};
  const float* As[3] = {(const float*)d_in[3], (const float*)d_in[7],  (const float*)d_in[11]};
  const float* Ad[3] = {(const float*)d_in[4], (const float*)d_in[8],  (const float*)d_in[12]};
  const float* Bi[3] = {(const float*)d_in[5], (const float*)d_in[9],  (const float*)d_in[13]};

  float* ws     = (float*)d_ws;
  float* bufH   = ws;                                   // N*96  (post-GEMM features)
  float* bufAcc = bufH   + (size_t)N_NODES * FDIM;      // N*96  (aggregate / next input)
  float* es     = bufAcc + (size_t)N_NODES * FDIM;      // N*3
  float* ed     = es     + (size_t)N_NODES * HEADS;     // N*3
  float* amax   = ed     + (size_t)N_NODES * HEADS;     // N*3
  float* den    = amax   + (size_t)N_NODES * HEADS;     // N*3
  float* alpha  = den    + (size_t)N_NODES * HEADS;     // E_TOT*3

  const int gemmTiles   = (N_NODES / 16) * (FDIM / 16); // 3125*6
  const int gemmBlocks  = (gemmTiles + 7) / 8;          // 8 waves / 256-thread block
  const int nhBlocks    = (N_NODES * HEADS + 255) / 256;
  const int nfBlocks    = (N_NODES * FDIM  + 255) / 256;
  const int eBlocks     = (E_TOT + 255) / 256;
  const int eWaveBlocks = (E_TOT + 7) / 8;              // one wave per edge

  const float* feat = x;
  for (int layer = 0; layer < 3; ++layer) {
    int K = (layer == 0) ? 128 : FDIM;
    gemm_wmma<<<gemmBlocks, 256, 0, stream>>>(feat, W[layer], bufH, N_NODES, K, FDIM);
    attn_scores<<<nhBlocks, 256, 0, stream>>>(bufH, As[layer], Ad[layer], es, ed);
    seg_init<<<nfBlocks, 256, 0, stream>>>(bufAcc, amax, den);
    edge_alpha_max<<<eBlocks, 256, 0, stream>>>(ei, es, ed, alpha, amax);
    edge_exp_sum<<<eBlocks, 256, 0, stream>>>(ei, amax, alpha, den);
    edge_aggregate<<<eWaveBlocks, 256, 0, stream>>>(ei, bufH, alpha, den, bufAcc);
    if (layer < 2) {
      bias_relu<<<nfBlocks, 256, 0, stream>>>(bufAcc, Bi[layer]);
      feat = bufAcc;
    } else {
      mean_bias_sigmoid<<<(N_NODES * CH + 255) / 256, 256, 0, stream>>>(
          bufAcc, Bi[2], (float*)d_out);
    }
  }
}